// OVDBaseClassifier_16355235463445
// MI455X (gfx1250) — compile-verified
//
#include <hip/hip_runtime.h>
#include <hip/hip_bf16.h>
#include <math.h>

typedef float v2f __attribute__((ext_vector_type(2)));
typedef float v8f __attribute__((ext_vector_type(8)));

#define KK 1849
#define DD 1024
#define NN 80
#define PP 43
#define SS 602
#define BB 4
#define EPSF 1e-8f

// ---------------------------------------------------------------------------
// Kernel 1: embedding norms  e_norm[n] = ||emb[n,:]||
// ---------------------------------------------------------------------------
__global__ __launch_bounds__(128) void ovd_enorm_kernel(const float* __restrict__ emb,
                                                        float* __restrict__ enorm) {
    __shared__ float red[128];
    const int n = blockIdx.x;
    const float* row = emb + (size_t)n * DD;
    float s = 0.f;
    for (int d = threadIdx.x; d < DD; d += 128) {
        float v = row[d];
        s += v * v;
    }
    red[threadIdx.x] = s;
    __syncthreads();
    for (int off = 64; off > 0; off >>= 1) {
        if (threadIdx.x < off) red[threadIdx.x] += red[threadIdx.x + off];
        __syncthreads();
    }
    if (threadIdx.x == 0) enorm[n] = sqrtf(red[0]);
}

// ---------------------------------------------------------------------------
// Kernel 2: bicubic tap table (43 -> 602, align_corners=False, a = -0.75).
// Stores pre-clamped source indices (int4) and tap weights (float4).
// Identical for H and W directions.
// ---------------------------------------------------------------------------
__global__ __launch_bounds__(256) void ovd_wtab_kernel(int4* __restrict__ idxTab,
                                                       float4* __restrict__ wtTab) {
    const int i = blockIdx.x * blockDim.x + threadIdx.x;
    if (i >= SS) return;
    const float A = -0.75f;
    const float scale = (float)PP / (float)SS;
    float src = ((float)i + 0.5f) * scale - 0.5f;
    float bf = floorf(src);
    float t  = src - bf;
    int   b  = (int)bf;

    float x0 = t + 1.0f;                                     // tap -1, x in [1,2)
    float w0 = A * (((x0 - 5.0f) * x0 + 8.0f) * x0 - 4.0f);
    float w1 = ((A + 2.0f) * t - (A + 3.0f)) * t * t + 1.0f; // tap 0,  x in [0,1)
    float u  = 1.0f - t;                                     // tap 1
    float w2 = ((A + 2.0f) * u - (A + 3.0f)) * u * u + 1.0f;
    float x3 = 2.0f - t;                                     // tap 2, x in (1,2]
    float w3 = A * (((x3 - 5.0f) * x3 + 8.0f) * x3 - 4.0f);

    int i0 = min(max(b - 1, 0), PP - 1);
    int i1 = min(max(b    , 0), PP - 1);
    int i2 = min(max(b + 1, 0), PP - 1);
    int i3 = min(max(b + 2, 0), PP - 1);

    idxTab[i] = make_int4(i0, i1, i2, i3);
    wtTab[i]  = make_float4(w0, w1, w2, w3);
}

// ---------------------------------------------------------------------------
// Kernel 3: cosine-similarity GEMM via V_WMMA_F32_16X16X4_F32.
//   cosim[b,n,k] = <feats[b,k,:], emb[n,:]> / (|feats[b,k]| * |emb[n]| + eps)
// Block = 160 threads = 5 waves; wave w owns N-tile [16w, 16w+16).
// Grid = (116 M-tiles, 4 batches). f_norm is accumulated from the A-fragment
// loads (each element appears exactly once across the 32-lane fragment).
// ---------------------------------------------------------------------------
__global__ __launch_bounds__(160) void ovd_cosim_kernel(const float* __restrict__ feats,
                                                        const float* __restrict__ emb,
                                                        const float* __restrict__ enorm,
                                                        float* __restrict__ cosim) {
    const int mt   = blockIdx.x;          // 0..115
    const int b    = blockIdx.y;          // 0..3
    const int wave = threadIdx.x >> 5;    // 0..4 -> N tile
    const int lane = threadIdx.x & 31;
    const int half = lane >> 4;           // 0: K taps {0,1}, 1: K taps {2,3}
    const int lm   = lane & 15;

    const int m0 = mt * 16;
    const int n0 = wave * 16;

    int rowA = m0 + lm;
    if (rowA > KK - 1) rowA = KK - 1;     // clamp tail tile (stores predicated below)

    const float* aptr = feats + ((size_t)b * KK + rowA) * DD + 2 * half;
    const float* bptr = emb + (size_t)(n0 + lm) * DD + 2 * half;

    v8f acc = {};
    float fsq = 0.f;

#pragma unroll 8
    for (int d = 0; d < DD; d += 4) {
        float2 a2 = *(const float2*)(aptr + d);
        float2 b2 = *(const float2*)(bptr + d);
        v2f av = {a2.x, a2.y};
        v2f bv = {b2.x, b2.y};
        fsq += a2.x * a2.x + a2.y * a2.y;
        // 8 args: (neg_a, A, neg_b, B, c_mod, C, reuse_a, reuse_b)
        acc = __builtin_amdgcn_wmma_f32_16x16x4_f32(
            false, av, false, bv, (short)0, acc, false, false);
    }

    // f_norm: lane l holds partial sum-of-squares of row (l&15), split over halves.
    float fnorm = sqrtf(fsq + __shfl_xor(fsq, 16, 32));  // valid per (lane&15)
    float en = enorm[n0 + lm];

    const int n = n0 + lm;
#pragma unroll
    for (int r = 0; r < 8; ++r) {
        int row = r + 8 * half;           // C layout: vgpr r -> M = r (+8 for hi half)
        int k = m0 + row;
        float frow = __shfl(fnorm, row, 32);
        float v = acc[r] / (frow * en + EPSF);
        if (k < KK)
            cosim[((size_t)b * NN + n) * (size_t)KK + k] = v;
    }
}

// ---------------------------------------------------------------------------
// Kernel 4: separable 4-tap bicubic upsample 43x43 -> 602x602.
// One block per (bn, H): vertical pass -> 43-wide LDS row, then 602 outputs.
// Output stores are non-temporal (TH=NT): 464 MB streamed once, keeping the
// hot working set (cosim tile re-read 602x, tap tables) resident in L2.
// ---------------------------------------------------------------------------
__global__ __launch_bounds__(256) void ovd_upsample_kernel(const float* __restrict__ cosim,
                                                           const int4* __restrict__ idxTab,
                                                           const float4* __restrict__ wtTab,
                                                           float* __restrict__ out) {
    const int H  = blockIdx.x;            // 0..601
    const int bn = blockIdx.y;            // 0..319
    const float* tile = cosim + (size_t)bn * KK;

    __shared__ float rowH[PP];

    const int4   ih = idxTab[H];
    const float4 wh = wtTab[H];

    const int t = threadIdx.x;
    if (t < PP) {
        rowH[t] = wh.x * tile[ih.x * PP + t] +
                  wh.y * tile[ih.y * PP + t] +
                  wh.z * tile[ih.z * PP + t] +
                  wh.w * tile[ih.w * PP + t];
    }
    __syncthreads();

    const size_t obase = ((size_t)bn * SS + H) * (size_t)SS;
    for (int W = t; W < SS; W += 256) {
        const int4   iw = idxTab[W];
        const float4 ww = wtTab[W];
        float v = ww.x * rowH[iw.x] +
                  ww.y * rowH[iw.y] +
                  ww.z * rowH[iw.z] +
                  ww.w * rowH[iw.w];
        __builtin_nontemporal_store(v, &out[obase + W]);
    }
}

// ---------------------------------------------------------------------------
// Host-side launcher
// ---------------------------------------------------------------------------
extern "C" void kernel_launch(void* const* d_in, const int* in_sizes, int n_in,
                              void* d_out, int out_size, void* d_ws, size_t ws_size,
                              hipStream_t stream) {
    (void)in_sizes; (void)n_in; (void)out_size; (void)ws_size;

    const float* feats = (const float*)d_in[0];   // [4, 1849, 1024] f32
    const float* emb   = (const float*)d_in[1];   // [80, 1024] f32
    float* out = (float*)d_out;                   // [4, 80, 602, 602] f32

    // Workspace carve-up (all 16B aligned):
    //   [0,    512)   : e_norm      (80 f32)
    //   [512,  10752) : idxTab      (602 int4)
    //   [10752,20992) : wtTab       (602 float4)
    //   [20992, ...)  : cosim       (4*80*1849 f32 ~ 2.37 MB)
    char*   wsb    = (char*)d_ws;
    float*  enorm  = (float*)(wsb);
    int4*   idxTab = (int4*)(wsb + 512);
    float4* wtTab  = (float4*)(wsb + 512 + 10240);
    float*  cosim  = (float*)(wsb + 512 + 10240 + 10240);

    ovd_enorm_kernel<<<NN, 128, 0, stream>>>(emb, enorm);
    ovd_wtab_kernel<<<(SS + 255) / 256, 256, 0, stream>>>(idxTab, wtTab);

    // 116 M-tiles of 16 cover K=1849; 5 waves cover N=80.
    ovd_cosim_kernel<<<dim3(116, BB), 160, 0, stream>>>(feats, emb, enorm, cosim);

    ovd_upsample_kernel<<<dim3(SS, BB * NN), 256, 0, stream>>>(cosim, idxTab, wtTab, out);
}